// UOT_60619168416427
// MI455X (gfx1250) — compile-verified
//
#include <hip/hip_runtime.h>
#include <stdint.h>
#include <math.h>

// Problem constants (match reference setup_inputs)
#define BB   64
#define NN   1024
#define DD   512
#define NUMK 4
#define EPSQ 1e-8f

#define TN   16              // rows per LDS tile in the column kernel
#define NBUF 4               // LDS tile buffers (prefetch depth 3)

typedef uint32_t u32;
typedef uint64_t u64;
typedef u32 v4u_t __attribute__((ext_vector_type(4)));
typedef int v8i_t __attribute__((ext_vector_type(8)));
typedef int v4i_t __attribute__((ext_vector_type(4)));

#if __has_builtin(__builtin_amdgcn_tensor_load_to_lds)
#define USE_TDM 1
#else
#define USE_TDM 0
#endif

// ---------------------------------------------------------------------------
// wave32 shuffle reductions (CDNA5 is wave32-only)
// ---------------------------------------------------------------------------
__device__ __forceinline__ float wredMax(float v) {
#pragma unroll
  for (int o = 16; o > 0; o >>= 1) v = fmaxf(v, __shfl_xor(v, o, 32));
  return v;
}
__device__ __forceinline__ float wredMin(float v) {
#pragma unroll
  for (int o = 16; o > 0; o >>= 1) v = fminf(v, __shfl_xor(v, o, 32));
  return v;
}
__device__ __forceinline__ float wredSum(float v) {
#pragma unroll
  for (int o = 16; o > 0; o >>= 1) v += __shfl_xor(v, o, 32);
  return v;
}

__device__ __forceinline__ float bredMax(float v, float* red) {
  int w = threadIdx.x >> 5, l = threadIdx.x & 31, nw = blockDim.x >> 5;
  v = wredMax(v);
  if (!l) red[w] = v;
  __syncthreads();
  float r = -INFINITY;
  for (int i = 0; i < nw; ++i) r = fmaxf(r, red[i]);
  __syncthreads();
  return r;
}
__device__ __forceinline__ float bredMin(float v, float* red) {
  int w = threadIdx.x >> 5, l = threadIdx.x & 31, nw = blockDim.x >> 5;
  v = wredMin(v);
  if (!l) red[w] = v;
  __syncthreads();
  float r = INFINITY;
  for (int i = 0; i < nw; ++i) r = fminf(r, red[i]);
  __syncthreads();
  return r;
}
__device__ __forceinline__ float bredSum(float v, float* red) {
  int w = threadIdx.x >> 5, l = threadIdx.x & 31, nw = blockDim.x >> 5;
  v = wredSum(v);
  if (!l) red[w] = v;
  __syncthreads();
  float r = 0.f;
  for (int i = 0; i < nw; ++i) r += red[i];
  __syncthreads();
  return r;
}

// ---------------------------------------------------------------------------
// Init kernels: build log_s, z and the small marginal arrays every call
// ---------------------------------------------------------------------------
__global__ void k_init_small(const float* __restrict__ p0,
                             const float* __restrict__ q0,
                             const float* __restrict__ mask,
                             float* log_p0, float* log_mu, float* z1,
                             float* log_q0, float* log_eta, float* z2) {
  int i = blockIdx.x * blockDim.x + threadIdx.x;   // [0, BB*NN)
  if (i < BB * DD) {
    float lp = __logf(p0[i]);
    log_p0[i] = lp;
    log_mu[i] = lp;
    z1[i] = 0.f;
  }
  float lq = __logf(q0[i] + EPSQ) * mask[i];
  log_q0[i] = lq;
  log_eta[i] = lq;
  z2[i] = 0.f;
}

__global__ void k_init_big(const float* __restrict__ p0,
                           const float* __restrict__ q0,
                           const float* __restrict__ mask,
                           float* __restrict__ log_s,
                           float* __restrict__ z) {
  int r = blockIdx.x * 2 + (threadIdx.x >> 7);     // row in [0, BB*NN)
  int q = threadIdx.x & 127;                       // float4 index within row
  int b = r >> 10;                                 // NN = 1024
  float mk = mask[r];
  float lq = __logf(q0[r] + EPSQ) * mk;
  const float4* p4 = (const float4*)(p0 + (size_t)b * DD);
  float4 pv = p4[q];
  float4 v;
  v.x = (lq + __logf(pv.x)) * mk;
  v.y = (lq + __logf(pv.y)) * mk;
  v.z = (lq + __logf(pv.z)) * mk;
  v.w = (lq + __logf(pv.w)) * mk;
  size_t i4 = (size_t)r * (DD / 4) + q;
  ((float4*)log_s)[i4] = v;
  ((float4*)z)[i4] = make_float4(0.f, 0.f, 0.f, 0.f);
}

// ---------------------------------------------------------------------------
// t-update: one wave per (b,n) row. LSE over D held entirely in registers.
// Writes log_t (into d_out as scratch) and tsum = sum_d exp(log_t)*mask,
// or, on the final iteration, writes exp(log_t)*mask directly to d_out.
// ---------------------------------------------------------------------------
__global__ void k_t(const float* __restrict__ x, const float* __restrict__ z,
                    const float* __restrict__ log_s,
                    const float* __restrict__ log_eta,
                    const float* __restrict__ mask,
                    float* __restrict__ out,      // log_t or exp(log_t)*mask
                    float* __restrict__ tsum,
                    const float* __restrict__ rho, int k, int writeExp) {
  int w = threadIdx.x >> 5, lane = threadIdx.x & 31;
  int r = blockIdx.x * 8 + w;                      // row, 8 waves per block
  float rn = rho[k];
  float invr = 1.0f / rn;
  float mk = mask[r];
  float leta = log_eta[r];
  size_t base4 = (size_t)r * (DD / 4);
  const float4* x4 = (const float4*)x + base4;
  const float4* z4 = (const float4*)z + base4;
  const float4* s4 = (const float4*)log_s + base4;

  float4 ya[4];
  float m = -INFINITY;
#pragma unroll
  for (int c = 0; c < 4; ++c) {
    int p = c * 32 + lane;
    float4 xv = x4[p], zv = z4[p], sv = s4[p];
    float4 y;
    y.x = (xv.x - zv.x) * invr + sv.x;
    y.y = (xv.y - zv.y) * invr + sv.y;
    y.z = (xv.z - zv.z) * invr + sv.z;
    y.w = (xv.w - zv.w) * invr + sv.w;
    ya[c] = y;
    m = fmaxf(m, fmaxf(fmaxf(y.x, y.y), fmaxf(y.z, y.w)));
  }
  m = wredMax(m);
  float s = 0.f;
#pragma unroll
  for (int c = 0; c < 4; ++c) {
    s += __expf(ya[c].x - m) + __expf(ya[c].y - m) +
         __expf(ya[c].z - m) + __expf(ya[c].w - m);
  }
  s = wredSum(s);
  float lse = m + __logf(s);
  float coef = mk * (leta - lse);                  // log_t = coef + y
  float4* o4 = (float4*)out + base4;
  float tacc = 0.f;
#pragma unroll
  for (int c = 0; c < 4; ++c) {
    int p = c * 32 + lane;
    float4 lt;
    lt.x = coef + ya[c].x; lt.y = coef + ya[c].y;
    lt.z = coef + ya[c].z; lt.w = coef + ya[c].w;
    if (writeExp) {
      float4 t;
      t.x = __expf(lt.x) * mk; t.y = __expf(lt.y) * mk;
      t.z = __expf(lt.z) * mk; t.w = __expf(lt.w) * mk;
      o4[p] = t;
    } else {
      o4[p] = lt;
      tacc += (__expf(lt.x) + __expf(lt.y) + __expf(lt.z) + __expf(lt.w)) * mk;
    }
  }
  if (!writeExp) {
    tacc = wredSum(tacc);
    if (lane == 0) tsum[r] = tacc;
  }
}

// ---------------------------------------------------------------------------
// Column pass tile streaming: TDM (tensor_load_to_lds) DMAs a full 16x128 f32
// tile per descriptor, issued by wave 0, tracked with TENSORcnt. Fallback:
// per-lane global_load_async_to_lds_b128 (ASYNCcnt). 4 buffers, depth-3
// prefetch => ~12 MB in flight across the grid at the HBM roofline.
// ---------------------------------------------------------------------------
#if USE_TDM
__device__ __forceinline__ void tdm_tile_load(u32 lds_addr, const float* g) {
  u64 ga = (u64)(uintptr_t)g;
  // D# group0: count=1 | lds_addr | global_addr[56:0] | type=2
  v4u_t g0 = {1u, lds_addr, (u32)ga,
              ((u32)((ga >> 32) & 0x01FFFFFFu)) | (2u << 30)};
  // D# group1: data_size=4B; tensor_dim0=DD; tensor_dim1=NN; tile_dim0=128;
  //            tile_dim1=TN; tensor_dim0_stride=DD
  v8i_t g1 = {(int)(2u << 16),        // wg_mask=0, data_size=2(4B)
              (int)((u32)DD << 16),   // tensor_dim0[15:0] << 16
              (int)((u32)NN << 16),   // tensor_dim0[31:16]=0 | tensor_dim1[15:0]
              (int)(128u << 16),      // tensor_dim1[31:16]=0 | tile_dim0=128
              TN,                     // tile_dim1=TN, tile_dim2=0
              DD,                     // tensor_dim0_stride[31:0]
              0, 0};
  v4i_t z4 = {0, 0, 0, 0};            // 2-D tensor: groups 2/3 unused
  v8i_t z8 = {0, 0, 0, 0, 0, 0, 0, 0};
  // amdgpu-toolchain (clang-23) 6-arg form:
  // (uint32x4 g0, int32x8 g1, int32x4, int32x4, int32x8, i32 cpol)
  __builtin_amdgcn_tensor_load_to_lds(g0, g1, z4, z4, z8, 0);
}
#endif

template <typename F>
__device__ __forceinline__ void staged_pass(const float* zBase,
                                            const float* tBase, int dbase,
                                            float* smem, F f) {
  const int tid = threadIdx.x;
  const u32 smemBase = (u32)(uintptr_t)smem;  // low 32 bits = LDS byte address
  const int NT = NN / TN;

  auto issue = [&](int tile) {
    int buf = tile & (NBUF - 1);
    u32 lz = smemBase + (u32)((buf * 2 + 0) * (TN * 128) * 4);
    u32 lt = smemBase + (u32)((buf * 2 + 1) * (TN * 128) * 4);
#if USE_TDM
    if (tid < 32) {
      size_t go = (size_t)(tile * TN) * DD + (size_t)dbase;
      tdm_tile_load(lz, zBase + go);
      tdm_tile_load(lt, tBase + go);
    }
#else
    const int w = tid >> 5, lane = tid & 31;
    const u64 zb = (u64)(uintptr_t)zBase;
    const u64 tb = (u64)(uintptr_t)tBase;
#pragma unroll
    for (int j = 0; j < 4; ++j) {
      int i = w * 4 + j;  // each wave DMAs 4 full 128-float rows per array
      u32 voff = (u32)(((tile * TN + i) * DD + dbase) * 4) + (lane << 4);
      u32 lo = (u32)(i * 512) + (lane << 4);
      asm volatile("global_load_async_to_lds_b128 %0, %1, %2"
                   :: "v"(lz + lo), "v"(voff), "s"(zb) : "memory");
      asm volatile("global_load_async_to_lds_b128 %0, %1, %2"
                   :: "v"(lt + lo), "v"(voff), "s"(tb) : "memory");
    }
#endif
  };

  auto wait_tiles = [&](int ahead) {  // wait until tile t has fully landed
#if USE_TDM
    if (tid < 32) {
      if (ahead >= 3)      __builtin_amdgcn_s_wait_tensorcnt(6);
      else if (ahead == 2) __builtin_amdgcn_s_wait_tensorcnt(4);
      else if (ahead == 1) __builtin_amdgcn_s_wait_tensorcnt(2);
      else                 __builtin_amdgcn_s_wait_tensorcnt(0);
    }
#else
    if (ahead >= 3)      asm volatile("s_wait_asynccnt 24" ::: "memory");
    else if (ahead == 2) asm volatile("s_wait_asynccnt 16" ::: "memory");
    else if (ahead == 1) asm volatile("s_wait_asynccnt 8" ::: "memory");
    else                 asm volatile("s_wait_asynccnt 0" ::: "memory");
#endif
  };

  issue(0); issue(1); issue(2);                    // NT = 64 >= 3
  for (int t = 0; t < NT; ++t) {
    if (t + 3 < NT) issue(t + 3);                  // overwrites buf (t-1)&3
    int ahead = NT - 1 - t;
    if (ahead > 3) ahead = 3;
    wait_tiles(ahead);
    __syncthreads();                               // tile t visible to block
    int buf = t & (NBUF - 1);
    const float* zt = smem + (buf * 2 + 0) * (TN * 128);
    const float* tt = smem + (buf * 2 + 1) * (TN * 128);
    int n0 = t * TN;
#pragma unroll
    for (int i = 0; i < TN; ++i) f(n0 + i, zt[i * 128 + tid], tt[i * 128 + tid]);
    __syncthreads();                               // guard buffer reuse
  }
}

// ---------------------------------------------------------------------------
// Column pass: one block per (b, 128-column slice). Two streamed passes:
//  A) online LSE over N (exact for binary mask: masked-out terms add
//     exp(0)*0 = 0 independent of the running max, so rescale is exact)
//  B) log_s, s, t, z-update, column sum of s
// ---------------------------------------------------------------------------
__global__ void k_cols(const float* __restrict__ log_t,   // lives in d_out
                       float* __restrict__ z, float* __restrict__ log_s,
                       const float* __restrict__ log_mu,
                       const float* __restrict__ mask,
                       float* __restrict__ ssum,
                       const float* __restrict__ a1,
                       const float* __restrict__ rho, int k) {
  __shared__ float smem[NBUF * 2 * TN * 128];             // 64 KB of 320 KB
  int b = blockIdx.x >> 2;                                // DD/128 = 4 slices
  int dbase = (blockIdx.x & 3) * 128;
  int tid = threadIdx.x;
  int d = dbase + tid;
  float rn = rho[k], a1k = a1[k];
  float inv = 1.0f / (a1k + rn);
  size_t colBase = (size_t)b * NN * DD;
  const float* zB = z + colBase;
  const float* tB = log_t + colBase;
  const float* mrow = mask + (size_t)b * NN;
  float lmu = log_mu[b * DD + d];

  // Pass A: online masked LSE of y = (z + rn*log_t)/(a1+rn) over n
  float m = -INFINITY, ssq = 0.f;
  staged_pass(zB, tB, dbase, smem, [&](int n, float zv, float tv) {
    float mk = mrow[n];
    float y = (zv + rn * tv) * inv;
    if (y > m) {
      ssq = ssq * __expf(m - y) + mk;   // exp((y-y)*mk)*mk == mk
      m = y;
    } else {
      ssq += __expf((y - m) * mk) * mk;
    }
  });
  float cs = lmu - __logf(ssq) - m;     // log_mu - lse - ymax
  // Pass B: log_s, s, t, z-update, column sum of s
  float sacc = 0.f;
  staged_pass(zB, tB, dbase, smem, [&](int n, float zv, float tv) {
    float mk = mrow[n];
    float y = (zv + rn * tv) * inv;
    float ls = mk * cs + y;
    float sv = __expf(ls) * mk;
    float te = __expf(tv) * mk;
    size_t idx = colBase + (size_t)n * DD + d;
    log_s[idx] = ls;
    z[idx] = zv + rn * (te - sv);
    sacc += sv;
  });
  ssum[b * DD + d] = sacc;
}

// ---------------------------------------------------------------------------
// mu-update + z1 dual update: one block per b, D=512 elems (2 per thread)
// ---------------------------------------------------------------------------
__global__ void k_mu(float* __restrict__ log_mu, const float* __restrict__ log_p0,
                     float* __restrict__ z1, const float* __restrict__ ssum,
                     const float* __restrict__ a2, const float* __restrict__ rho,
                     int k) {
  __shared__ float red[8];
  int b = blockIdx.x, tid = threadIdx.x;
  float rn = rho[k], a2k = a2[k];
  float inv = 1.0f / (rn + a2k);
  int i0 = b * DD + tid, i1 = i0 + 256;
  float z1v0 = z1[i0], z1v1 = z1[i1];
  float y0 = (rn * log_mu[i0] + a2k * log_p0[i0] - z1v0) * inv;
  float y1 = (rn * log_mu[i1] + a2k * log_p0[i1] - z1v1) * inv;
  float m = bredMax(fmaxf(y0, y1), red);
  float s = bredSum(__expf(y0 - m) + __expf(y1 - m), red);
  float lse = m + __logf(s);
  float n0 = y0 - lse, n1 = y1 - lse;
  log_mu[i0] = n0;
  log_mu[i1] = n1;
  z1[i0] = z1v0 + rn * (__expf(n0) - ssum[i0]);
  z1[i1] = z1v1 + rn * (__expf(n1) - ssum[i1]);
}

// ---------------------------------------------------------------------------
// eta-update + z2 dual update: one block per b, N=1024 elems (4 per thread),
// exact masked column LSE (min, masked max, masked sumexp)
// ---------------------------------------------------------------------------
__global__ void k_eta(float* __restrict__ log_eta, const float* __restrict__ log_q0,
                      float* __restrict__ z2, const float* __restrict__ tsum,
                      const float* __restrict__ mask,
                      const float* __restrict__ a3, const float* __restrict__ rho,
                      int k) {
  __shared__ float red[8];
  int b = blockIdx.x, tid = threadIdx.x;
  float rn = rho[k], a3k = a3[k];
  float inv = 1.0f / (rn + a3k);
  float y[4], mk[4], z2v[4];
  int idx[4];
#pragma unroll
  for (int j = 0; j < 4; ++j) {
    idx[j] = b * NN + tid + j * 256;
    mk[j] = mask[idx[j]];
    z2v[j] = z2[idx[j]];
    y[j] = (rn * log_eta[idx[j]] + a3k * log_q0[idx[j]] - z2v[j]) * inv;
  }
  float vmin = fminf(fminf(y[0], y[1]), fminf(y[2], y[3]));
  float ymin = bredMin(vmin, red);
  float vmax = -INFINITY;
#pragma unroll
  for (int j = 0; j < 4; ++j) vmax = fmaxf(vmax, ymin - mk[j] * ymin + y[j]);
  float ymax = bredMax(vmax, red);
  float sl = 0.f;
#pragma unroll
  for (int j = 0; j < 4; ++j) sl += __expf((y[j] - ymax) * mk[j]) * mk[j];
  float lse = __logf(bredSum(sl, red));
#pragma unroll
  for (int j = 0; j < 4; ++j) {
    float le = (y[j] - lse - ymax) * mk[j];
    log_eta[idx[j]] = le;
    z2[idx[j]] = z2v[j] + rn * (__expf(le) * mk[j] - tsum[idx[j]]) * mk[j];
  }
}

// ---------------------------------------------------------------------------
// Host launcher
// ---------------------------------------------------------------------------
extern "C" void kernel_launch(void* const* d_in, const int* in_sizes, int n_in,
                              void* d_out, int out_size, void* d_ws,
                              size_t ws_size, hipStream_t stream) {
  (void)in_sizes; (void)n_in; (void)out_size; (void)ws_size;
  const float* x    = (const float*)d_in[0];
  const float* p0   = (const float*)d_in[1];
  const float* q0   = (const float*)d_in[2];
  const float* a1   = (const float*)d_in[3];
  const float* a2   = (const float*)d_in[4];
  const float* a3   = (const float*)d_in[5];
  const float* rho  = (const float*)d_in[6];
  const float* mask = (const float*)d_in[7];
  float* out = (float*)d_out;

  const size_t SZ = (size_t)BB * NN * DD;   // 33,554,432 floats
  const size_t BD = (size_t)BB * DD;        // 32,768
  const size_t BN = (size_t)BB * NN;        // 65,536
  float* w = (float*)d_ws;
  float* log_s   = w;
  float* z       = w + SZ;
  float* sm      = w + 2 * SZ;
  float* log_p0  = sm;
  float* log_mu  = sm + BD;
  float* z1      = sm + 2 * BD;
  float* ssum    = sm + 3 * BD;
  float* log_q0  = sm + 4 * BD;
  float* log_eta = sm + 4 * BD + BN;
  float* z2      = sm + 4 * BD + 2 * BN;
  float* tsum    = sm + 4 * BD + 3 * BN;

  k_init_small<<<BN / 256, 256, 0, stream>>>(p0, q0, mask, log_p0, log_mu, z1,
                                             log_q0, log_eta, z2);
  k_init_big<<<BN / 2, 256, 0, stream>>>(p0, q0, mask, log_s, z);

  for (int k = 0; k < NUMK; ++k) {
    int fin = (k == NUMK - 1);
    k_t<<<BN / 8, 256, 0, stream>>>(x, z, log_s, log_eta, mask, out, tsum, rho,
                                    k, fin);
    if (fin) break;   // only the t-update of the last iteration affects output
    k_cols<<<BB * (DD / 128), 128, 0, stream>>>(out, z, log_s, log_mu, mask,
                                                ssum, a1, rho, k);
    k_mu<<<BB, 256, 0, stream>>>(log_mu, log_p0, z1, ssum, a2, rho, k);
    k_eta<<<BB, 256, 0, stream>>>(log_eta, log_q0, z2, tsum, mask, a3, rho, k);
  }
}